// WarpTorchOP_7206955122765
// MI455X (gfx1250) — compile-verified
//
#include <hip/hip_runtime.h>

// Homography warp + bilinear sample, B=8, C=3, 1024x1024 fp32.
// Bandwidth-bound (~200 MB total traffic): optimize the memory path, not FLOPs.
//  - src (96 MB) fits in MI455X's 192 MB L2 -> keep it cached.
//  - output is write-once -> non-temporal stores so it doesn't evict src.
//  - global_prefetch_b8 to launch all 3 channel-plane fetches early.
//  - H is wave-uniform (indexed by blockIdx) -> scalar loads, gathers use
//    SGPR-base + VGPR-offset addressing.

#define IMG_W   1024
#define IMG_H   1024
#define PLANE   (IMG_W * IMG_H)
#define SMALL_T 1e-7f
#define EPS_T   1e-6f

__global__ __launch_bounds__(256) void warp_bilinear_kernel(
    const float* __restrict__ src,   // [B,3,1024,1024]
    const float* __restrict__ Hm,    // [B,3,3]
    float* __restrict__ out)         // [B,3,1024,1024]
{
    const int b   = blockIdx.y;
    const int tid = blockIdx.x * 256 + threadIdx.x;   // pixel index in plane
    const int px  = tid & (IMG_W - 1);
    const int py  = tid >> 10;

    // Uniform per-block homography: compiler scalarizes these to s_load.
    const float* __restrict__ Hb = Hm + b * 9;
    const float h0 = Hb[0], h1 = Hb[1], h2 = Hb[2];
    const float h3 = Hb[3], h4 = Hb[4], h5 = Hb[5];
    const float h6 = Hb[6], h7 = Hb[7], h8 = Hb[8];

    const float fx = (float)px, fy = (float)py;
    float X = fmaf(h0, fx, fmaf(h1, fy, h2));
    float Y = fmaf(h3, fx, fmaf(h4, fy, h5));
    float T = fmaf(h6, fx, fmaf(h7, fy, h8));

    // T += EPS * (1 - (|T| >= SMALL))
    T += (__builtin_fabsf(T) >= SMALL_T) ? 0.0f : EPS_T;

    // One v_rcp_f32 instead of two IEEE div expansions.
    const float r = __builtin_amdgcn_rcpf(T);
    const float x = X * r;
    const float y = Y * r;

    const float fx0 = __builtin_floorf(x);
    const float fy0 = __builtin_floorf(y);
    int x0 = (int)fx0;
    int y0 = (int)fy0;
    int x1 = min(max(x0 + 1, 0), IMG_W - 1);
    int y1 = min(max(y0 + 1, 0), IMG_H - 1);
    x0 = min(max(x0, 0), IMG_W - 1);
    y0 = min(max(y0, 0), IMG_H - 1);

    const float x0f = (float)x0, x1f = (float)x1;
    const float y0f = (float)y0, y1f = (float)y1;
    const float wa = (x1f - x) * (y1f - y);   // (y0,x0)
    const float wb = (x1f - x) * (y - y0f);   // (y1,x0)
    const float wc = (x - x0f) * (y1f - y);   // (y0,x1)
    const float wd = (x - x0f) * (y - y0f);   // (y1,x1)

    const int i00 = y0 * IMG_W + x0;
    const int i10 = y1 * IMG_W + x0;
    const int i01 = y0 * IMG_W + x1;
    const int i11 = y1 * IMG_W + x1;

    const float* __restrict__ p0 = src + (size_t)(b * 3 + 0) * PLANE;
    const float* __restrict__ p1 = src + (size_t)(b * 3 + 1) * PLANE;
    const float* __restrict__ p2 = src + (size_t)(b * 3 + 2) * PLANE;

    // gfx1250 global_prefetch_b8: launch all three channel-plane lines
    // before the dependent gather/blend chain.
    __builtin_prefetch(p0 + i00, 0, 3);
    __builtin_prefetch(p1 + i00, 0, 3);
    __builtin_prefetch(p2 + i00, 0, 3);

    const float o0 = fmaf(wa, p0[i00], fmaf(wb, p0[i10], fmaf(wc, p0[i01], wd * p0[i11])));
    const float o1 = fmaf(wa, p1[i00], fmaf(wb, p1[i10], fmaf(wc, p1[i01], wd * p1[i11])));
    const float o2 = fmaf(wa, p2[i00], fmaf(wb, p2[i10], fmaf(wc, p2[i01], wd * p2[i11])));

    // Non-temporal stores: output is never re-read; keep the 96 MB src
    // working set resident in the 192 MB L2 instead.
    float* q = out + (size_t)(b * 3) * PLANE + tid;
    __builtin_nontemporal_store(o0, q);
    __builtin_nontemporal_store(o1, q + PLANE);
    __builtin_nontemporal_store(o2, q + 2 * PLANE);
}

extern "C" void kernel_launch(void* const* d_in, const int* in_sizes, int n_in,
                              void* d_out, int out_size, void* d_ws, size_t ws_size,
                              hipStream_t stream) {
    const float* src = (const float*)d_in[0];
    const float* Hm  = (const float*)d_in[1];
    float* out       = (float*)d_out;

    const int B = in_sizes[1] / 9;          // 8 for the reference shapes
    dim3 grid(PLANE / 256, B, 1);           // 4096 x 8 blocks, 256 thr (8 wave32)
    warp_bilinear_kernel<<<grid, 256, 0, stream>>>(src, Hm, out);
}